// Raymarcher_83494164234659
// MI455X (gfx1250) — compile-verified
//
#include <hip/hip_runtime.h>

typedef __attribute__((ext_vector_type(2))) float v2f;
typedef __attribute__((ext_vector_type(8))) float v8f;

#define NRAYS   16384
#define NSTEPS  64
#define KPRIM   32
#define DT      (1.0f/64.0f)

// Trilinear sample of primitive p's 4-channel 16^3 template (channel-major:
// c*4096 + z*256 + y*16 + x) at primitive-frame coords (y0->z, y1->y, y2->x),
// matching the reference's inside test and clamping. Accumulates into sr..sa.
__device__ __forceinline__ void sample_prim(const float* __restrict__ tp,
                                            float y0, float y1, float y2,
                                            float& sr, float& sg, float& sb, float& sa)
{
    if (!(fabsf(y0) <= 1.0f && fabsf(y1) <= 1.0f && fabsf(y2) <= 1.0f)) return;
    float g0 = (y0 + 1.0f) * 7.5f;
    float g1 = (y1 + 1.0f) * 7.5f;
    float g2 = (y2 + 1.0f) * 7.5f;
    int iz = (int)floorf(g0); iz = iz < 0 ? 0 : (iz > 14 ? 14 : iz);
    int iy = (int)floorf(g1); iy = iy < 0 ? 0 : (iy > 14 ? 14 : iy);
    int ix = (int)floorf(g2); ix = ix < 0 ? 0 : (ix > 14 ? 14 : ix);
    float fz = g0 - (float)iz; fz = fz < 0.f ? 0.f : (fz > 1.f ? 1.f : fz);
    float fy = g1 - (float)iy; fy = fy < 0.f ? 0.f : (fy > 1.f ? 1.f : fy);
    float fx = g2 - (float)ix; fx = fx < 0.f ? 0.f : (fx > 1.f ? 1.f : fx);

    const float* b0 = tp + iz * 256 + iy * 16 + ix;
    float w00 = (1.f - fz) * (1.f - fy);
    float w01 = (1.f - fz) * fy;
    float w10 = fz * (1.f - fy);
    float w11 = fz * fy;
    float x0 = 1.f - fx, x1 = fx;

    float acc[4];
#pragma unroll
    for (int c = 0; c < 4; ++c) {
        const float* bc = b0 + c * 4096;
        acc[c] = w00 * (x0 * bc[0]   + x1 * bc[1])
               + w01 * (x0 * bc[16]  + x1 * bc[17])
               + w10 * (x0 * bc[256] + x1 * bc[257])
               + w11 * (x0 * bc[272] + x1 * bc[273]);
    }
    sr += acc[0]; sg += acc[1]; sb += acc[2]; sa += acc[3];
}

// One wave (32 lanes) processes 16 rays; lanes r and r+16 co-own ray r.
// Per step, 8 x V_WMMA_F32_16X16X4_F32 compute yloc for all 32 primitives
// (padded to 4 rows each) x 16 rays. D layout: lane n holds rows 0-7 of
// column n (lanes 0-15) / rows 8-15 (lanes 16-31), so lane half r handles
// primitives {4w,4w+1} and lane half r+16 handles {4w+2,4w+3} in-lane.
__global__ __launch_bounds__(256) void raymarch_wmma(
    const float* __restrict__ raypos, const float* __restrict__ raydir,
    const float* __restrict__ tminmax, const float* __restrict__ primpos,
    const float* __restrict__ primrot, const float* __restrict__ primscale,
    const float* __restrict__ tmpl, float* __restrict__ out)
{
    const int lane = threadIdx.x & 31;
    const int wave = threadIdx.x >> 5;
    const int half = lane >> 4;      // 0: rows 0-7 / B rows 0,1   1: rows 8-15 / B rows 2,3
    const int l16  = lane & 15;
    const int ray  = blockIdx.x * 128 + wave * 16 + l16;

    // Warm L2 with this block's slice of the (2 MB, L2-resident) template.
    {
        const char* base = (const char*)tmpl;
        __builtin_prefetch(base + (size_t)blockIdx.x * 16384 + (size_t)threadIdx.x * 64, 0, 1);
    }

    // Build the 8 A operands (16x4 f32 each). Row m = l16 of WMMA w maps to
    // primitive p = 4w + (m>>2), component i = m&3 (i==3 -> zero pad row).
    // A layout: lanes 0-15 hold (K=0,K=1) in the two VGPRs, lanes 16-31 (K=2,K=3).
    v2f a[8];
    {
        const int m = l16;
        const int comp = m & 3;
#pragma unroll
        for (int w = 0; w < 8; ++w) {
            const int p = 4 * w + (m >> 2);
            float a0 = 0.f, a1 = 0.f;
            if (comp < 3) {
                float s  = primscale[p * 3 + comp];
                float r0 = primrot[p * 9 + comp * 3 + 0];
                float r1 = primrot[p * 9 + comp * 3 + 1];
                float r2 = primrot[p * 9 + comp * 3 + 2];
                float px = primpos[p * 3 + 0];
                float py = primpos[p * 3 + 1];
                float pz = primpos[p * 3 + 2];
                float bias = -s * (r0 * px + r1 * py + r2 * pz);
                if (half == 0) { a0 = s * r0; a1 = s * r1; }   // K=0,1
                else           { a0 = s * r2; a1 = bias;   }   // K=2,3
            }
            a[w].x = a0; a[w].y = a1;
        }
    }

    // Per-ray data (both lanes of a pair load the same ray).
    const float rpx = raypos[ray * 3 + 0];
    const float rpy = raypos[ray * 3 + 1];
    const float rpz = raypos[ray * 3 + 2];
    const float rdx = raydir[ray * 3 + 0];
    const float rdy = raydir[ray * 3 + 1];
    const float rdz = raydir[ray * 3 + 2];
    const float tmin = tminmax[ray * 2 + 0];
    const float tmax = tminmax[ray * 2 + 1];

    float rgbx = 0.f, rgby = 0.f, rgbz = 0.f, alpha = 0.f;

#pragma unroll 1
    for (int it = 0; it < NSTEPS; ++it) {
        const float t = tmin + (float)it * DT;

        // B operand (4x16 f32): row K striped across lanes within a VGPR.
        // lanes 0-15: (K=0,K=1)=(pos.x,pos.y); lanes 16-31: (K=2,K=3)=(pos.z,1).
        v2f b;
        if (half == 0) { b.x = rpx + rdx * t; b.y = rpy + rdy * t; }
        else           { b.x = rpz + rdz * t; b.y = 1.0f;          }

        float sr = 0.f, sg = 0.f, sb = 0.f, sa = 0.f;
        const bool active = (t < tmax) && (alpha < 1.0f);

#pragma unroll
        for (int w = 0; w < 8; ++w) {
            v8f c = {0.f, 0.f, 0.f, 0.f, 0.f, 0.f, 0.f, 0.f};
            // D = A x B : yloc for 4 primitives (rows) x 16 rays (cols)
            v8f d = __builtin_amdgcn_wmma_f32_16x16x4_f32(
                false, a[w], false, b, (short)0, c, false, false);
            if (active) {
                const int p0 = 4 * w + 2 * half;   // this lane half's 2 primitives
                sample_prim(tmpl + (size_t)p0 * 16384,       d[0], d[1], d[2], sr, sg, sb, sa);
                sample_prim(tmpl + (size_t)(p0 + 1) * 16384, d[4], d[5], d[6], sr, sg, sb, sa);
            }
        }

        // Merge the two half-lane partial sums for this ray (symmetric -> both
        // lanes of the pair hold the full sums and identical composite state).
        sr += __shfl_xor(sr, 16, 32);
        sg += __shfl_xor(sg, 16, 32);
        sb += __shfl_xor(sb, 16, 32);
        sa += __shfl_xor(sa, 16, 32);

        const float valid = (t < tmax) ? 1.0f : 0.0f;
        float na = alpha + sa * DT * valid;
        na = na < 0.f ? 0.f : (na > 1.f ? 1.f : na);
        const float contrib = na - alpha;
        rgbx += sr * contrib;
        rgby += sg * contrib;
        rgbz += sb * contrib;
        alpha = na;
    }

    if (half == 0) {
        // rayrgb (1,3,128,128), rayalpha (1,1,128,128), rayrgba (1,4,128,128), planar.
        out[0 * NRAYS + ray] = rgbx;
        out[1 * NRAYS + ray] = rgby;
        out[2 * NRAYS + ray] = rgbz;
        out[3 * NRAYS + ray] = alpha;
        out[4 * NRAYS + ray] = rgbx;
        out[5 * NRAYS + ray] = rgby;
        out[6 * NRAYS + ray] = rgbz;
        out[7 * NRAYS + ray] = alpha;
    }
}

extern "C" void kernel_launch(void* const* d_in, const int* in_sizes, int n_in,
                              void* d_out, int out_size, void* d_ws, size_t ws_size,
                              hipStream_t stream) {
    (void)in_sizes; (void)n_in; (void)out_size; (void)d_ws; (void)ws_size;
    const float* raypos    = (const float*)d_in[0];
    const float* raydir    = (const float*)d_in[1];
    const float* tminmax   = (const float*)d_in[2];
    const float* primpos   = (const float*)d_in[3];
    const float* primrot   = (const float*)d_in[4];
    const float* primscale = (const float*)d_in[5];
    const float* tmpl      = (const float*)d_in[6];
    float* out = (float*)d_out;

    // 16384 rays, 16 rays per wave, 8 waves (256 threads) per block -> 128 blocks.
    raymarch_wmma<<<dim3(NRAYS / 128), dim3(256), 0, stream>>>(
        raypos, raydir, tminmax, primpos, primrot, primscale, tmpl, out);
}